// Decoder_77455440216072
// MI455X (gfx1250) — compile-verified
//
#include <hip/hip_runtime.h>

typedef float v2f  __attribute__((ext_vector_type(2)));
typedef float v8f  __attribute__((ext_vector_type(8)));
typedef float f32x4 __attribute__((ext_vector_type(4)));

#define DM 256        // d_model
#define QKV 512       // H * DK = 32*16
#define NLAYERS 5

// Async copy of 16B/lane from global to LDS (ASYNCcnt-tracked).
__device__ __forceinline__ void async_b128(uint32_t lds_addr, const void* gptr) {
  asm volatile("global_load_async_to_lds_b128 %0, %1, off"
               :: "v"(lds_addr), "v"(gptr) : "memory");
}
__device__ __forceinline__ void wait_async0() {
  asm volatile("s_wait_asynccnt 0" ::: "memory");
}

// ---------------------------------------------------------------------------
// Tiled fp32 WMMA GEMM:  Y[M,Nout] = X[M,Kd] @ W[Nout,Kd]^T (+bias[n]) (+resid)
// 128 threads (4 waves); tile 64x64; double-buffered LDS, K-chunk 32, fed by
// global_load_async_to_lds_b128 overlapping WMMA on the other buffer.
// ---------------------------------------------------------------------------
__global__ __launch_bounds__(128) void gemm_wmma(
    const float* __restrict__ X, const float* __restrict__ W,
    const float* __restrict__ bias, const float* __restrict__ resid,
    float* __restrict__ Y, int M, int Kd, int Nout) {
  __shared__ float Xs[2][64][36];   // stride 36 words: 16B-aligned, bank-spread
  __shared__ float Ws[2][64][36];

  const int tid  = threadIdx.x;
  const int lane = tid & 31;
  const int wave = tid >> 5;
  const int lr   = lane & 15;
  const int hi   = lane >> 4;          // which 16-lane half
  const int m0   = blockIdx.x * 64;
  const int n0   = blockIdx.y * 64;

  v8f acc[4];
  #pragma unroll
  for (int t = 0; t < 4; ++t)
    #pragma unroll
    for (int i = 0; i < 8; ++i) acc[t][i] = 0.0f;

  // Issue async fills of one 64x32 tile pair into buffer `buf`.
  auto load_tiles = [&](int kb, int buf) {
    #pragma unroll
    for (int j = 0; j < 4; ++j) {
      int f   = tid + j * 128;          // 0..511 float4 slots per operand
      int row = f >> 3;                 // 8 float4 per 32-wide row
      int c4  = (f & 7) << 2;
      int gm  = m0 + row;
      if (gm >= M) gm = M - 1;          // clamp: garbage lands in dead C rows
      async_b128((uint32_t)(uintptr_t)&Xs[buf][row][c4],
                 X + (size_t)gm * Kd + kb + c4);
      async_b128((uint32_t)(uintptr_t)&Ws[buf][row][c4],
                 W + (size_t)(n0 + row) * Kd + kb + c4);
    }
  };

  int nbuf = 0;
  load_tiles(0, nbuf);
  wait_async0();
  __syncthreads();

  const int r0 = wave * 16;
  for (int kb = 0; kb < Kd; kb += 32) {
    const int cur = nbuf;
    nbuf ^= 1;
    if (kb + 32 < Kd) load_tiles(kb + 32, nbuf);  // prefetch next chunk

    #pragma unroll
    for (int k = 0; k < 32; k += 4) {
      // A fragment (16x4): lanes 0-15 -> K=k,k+1 ; lanes 16-31 -> K=k+2,k+3
      v2f a;
      a.x = Xs[cur][r0 + lr][k + 2 * hi];
      a.y = Xs[cur][r0 + lr][k + 2 * hi + 1];
      #pragma unroll
      for (int t = 0; t < 4; ++t) {
        // B fragment (4x16): B[k][n] = W[n][k] -> same addressing pattern
        v2f b;
        b.x = Ws[cur][t * 16 + lr][k + 2 * hi];
        b.y = Ws[cur][t * 16 + lr][k + 2 * hi + 1];
        acc[t] = __builtin_amdgcn_wmma_f32_16x16x4_f32(
            false, a, false, b, (short)0, acc[t], false, false);
      }
    }
    wait_async0();
    __syncthreads();
  }

  // C fragment store: elem r -> row (r + 8*hi), col (lr)
  #pragma unroll
  for (int t = 0; t < 4; ++t) {
    int n = n0 + t * 16 + lr;
    float bv = bias ? bias[n] : 0.0f;
    #pragma unroll
    for (int r = 0; r < 8; ++r) {
      int m = m0 + r0 + r + 8 * hi;
      if (m < M) {
        float val = acc[t][r] + bv;
        if (resid) val += resid[(size_t)m * Nout + n];
        Y[(size_t)m * Nout + n] = val;
      }
    }
  }
}

// ---------------------------------------------------------------------------
// Per-edge attention: one wave per edge (4 edges / 128-thread block).
// alpha[a][b] = sum_h Q[dst][h*16+a]*K[src][h*16+b] / 4   (16x16, contract 32)
// att = softmax_rows(alpha); msg[a][h] = sum_b att[a][b]*V[src][h*16+b]
// atomicAdd into attn[n][h*16+a]  (combined-heads layout, [N,512])
// ---------------------------------------------------------------------------
__global__ __launch_bounds__(128) void edge_attn(
    const float* __restrict__ Q, const float* __restrict__ K,
    const float* __restrict__ V, const int* __restrict__ ei,
    float* __restrict__ attn, int E) {
  __shared__ float qr[4][QKV];
  __shared__ float kr[4][QKV];
  __shared__ float vr[4][QKV];
  __shared__ float att[4][16][17];

  const int tid  = threadIdx.x;
  const int lane = tid & 31;
  const int w    = tid >> 5;
  const int lr   = lane & 15;
  const int hi   = lane >> 4;
  const int e    = blockIdx.x * 4 + w;
  const bool active = (e < E);
  const int srcn = active ? ei[e] : 0;
  const int dstn = active ? ei[E + e] : 0;

  // Stage Q[dst], K[src], V[src] rows (512 floats each) via async-to-LDS.
  {
    const f32x4* qs = (const f32x4*)(Q + (size_t)dstn * QKV);
    const f32x4* ks = (const f32x4*)(K + (size_t)srcn * QKV);
    const f32x4* vs = (const f32x4*)(V + (size_t)srcn * QKV);
    #pragma unroll
    for (int j = 0; j < 4; ++j) {
      int idx = lane + 32 * j;
      async_b128((uint32_t)(uintptr_t)&qr[w][idx * 4], qs + idx);
      async_b128((uint32_t)(uintptr_t)&kr[w][idx * 4], ks + idx);
      async_b128((uint32_t)(uintptr_t)&vr[w][idx * 4], vs + idx);
    }
  }
  wait_async0();
  __syncthreads();

  // alpha = A(16x32) * B(32x16); A[a][h]=Qrow[h*16+a], B[h][b]=Krow[h*16+b]
  v8f c;
  #pragma unroll
  for (int i = 0; i < 8; ++i) c[i] = 0.0f;
  #pragma unroll
  for (int h = 0; h < 32; h += 4) {
    int col = h + 2 * hi;
    v2f a, b;
    a.x = qr[w][col * 16 + lr];
    a.y = qr[w][(col + 1) * 16 + lr];
    b.x = kr[w][col * 16 + lr];
    b.y = kr[w][(col + 1) * 16 + lr];
    c = __builtin_amdgcn_wmma_f32_16x16x4_f32(false, a, false, b, (short)0, c,
                                              false, false);
  }

  // scale + row softmax (rows live across each 16-lane half)
  #pragma unroll
  for (int r = 0; r < 8; ++r) {
    float av = c[r] * 0.25f;  // 1/sqrt(16)
    float mx = av;
    #pragma unroll
    for (int msk = 8; msk >= 1; msk >>= 1) mx = fmaxf(mx, __shfl_xor(mx, msk));
    float ev = __expf(av - mx);
    float sm = ev;
    #pragma unroll
    for (int msk = 8; msk >= 1; msk >>= 1) sm += __shfl_xor(sm, msk);
    att[w][r + 8 * hi][lr] = ev / sm;   // C layout -> row-major via LDS
  }
  __syncthreads();

  // msg = att(16x16) * B(16x32); B[b][h] = Vrow[h*16+b]; two 16-col tiles
  #pragma unroll
  for (int t = 0; t < 2; ++t) {
    v8f m8;
    #pragma unroll
    for (int i = 0; i < 8; ++i) m8[i] = 0.0f;
    #pragma unroll
    for (int kk = 0; kk < 16; kk += 4) {
      int col = kk + 2 * hi;
      v2f a, b;
      a.x = att[w][lr][col];
      a.y = att[w][lr][col + 1];
      b.x = vr[w][(t * 16 + lr) * 16 + col];
      b.y = vr[w][(t * 16 + lr) * 16 + col + 1];
      m8 = __builtin_amdgcn_wmma_f32_16x16x4_f32(false, a, false, b, (short)0,
                                                 m8, false, false);
    }
    if (active) {
      float* base = attn + (size_t)dstn * QKV;
      #pragma unroll
      for (int r = 0; r < 8; ++r)
        atomicAdd(base + (t * 16 + lr) * 16 + r + 8 * hi, m8[r]);
    }
  }
}

// ---------------------------------------------------------------------------
// LayerNorm over D=256: out = LN(A (+B)) * g + b ; one block per row
// ---------------------------------------------------------------------------
__global__ __launch_bounds__(256) void layernorm256(
    const float* __restrict__ A, const float* __restrict__ B,
    const float* __restrict__ g, const float* __restrict__ bb,
    float* __restrict__ out) {
  __shared__ float red[256];
  const int row = blockIdx.x, t = threadIdx.x;
  float v = A[(size_t)row * DM + t];
  if (B) v += B[(size_t)row * DM + t];
  red[t] = v;
  __syncthreads();
  for (int s = 128; s > 0; s >>= 1) {
    if (t < s) red[t] += red[t + s];
    __syncthreads();
  }
  float mean = red[0] * (1.0f / DM);
  __syncthreads();
  float d = v - mean;
  red[t] = d * d;
  __syncthreads();
  for (int s = 128; s > 0; s >>= 1) {
    if (t < s) red[t] += red[t + s];
    __syncthreads();
  }
  float inv = rsqrtf(red[0] * (1.0f / DM) + 1e-5f);
  out[(size_t)row * DM + t] = d * inv * g[t] + bb[t];
}

__global__ void zerof(float* __restrict__ p, int n) {
  int i = blockIdx.x * blockDim.x + threadIdx.x;
  if (i < n) p[i] = 0.0f;
}

// ---------------------------------------------------------------------------
extern "C" void kernel_launch(void* const* d_in, const int* in_sizes, int n_in,
                              void* d_out, int out_size, void* d_ws,
                              size_t ws_size, hipStream_t stream) {
  const int*   ei    = (const int*)d_in[0];
  const float* x     = (const float*)d_in[1];
  const float* We    = (const float*)d_in[2];
  const float* Wq    = (const float*)d_in[3];
  const float* Wk    = (const float*)d_in[4];
  const float* Wv    = (const float*)d_in[5];
  const float* Wo    = (const float*)d_in[6];
  const float* bo    = (const float*)d_in[7];
  const float* ln_g  = (const float*)d_in[8];
  const float* ln_b  = (const float*)d_in[9];
  const float* mlp_W = (const float*)d_in[10];
  const float* mlp_b = (const float*)d_in[11];

  const int E = in_sizes[0] / 2;
  const int N = in_sizes[1] / DM;

  float* ws = (float*)d_ws;
  const size_t s256 = (size_t)N * DM;
  const size_t s512 = (size_t)N * QKV;
  float* xcur = ws;
  float* xe   = xcur + s256;
  float* qb   = xe + s256;
  float* kb   = qb + s512;
  float* vb   = kb + s512;
  float* ab   = vb + s512;   // attn accumulator [N,512]
  float* hb   = ab + s512;   // [N,256]

  hipMemcpyAsync(xcur, x, s256 * sizeof(float), hipMemcpyDeviceToDevice,
                 stream);

  const dim3 blk(128);
  auto gemm = [&](const float* X, const float* W, const float* bias,
                  const float* resid, float* Y, int M, int Kd, int Nout) {
    dim3 grid((M + 63) / 64, Nout / 64);
    gemm_wmma<<<grid, blk, 0, stream>>>(X, W, bias, resid, Y, M, Kd, Nout);
  };

  for (int l = 0; l < NLAYERS; ++l) {
    for (int s = 0; s < 2; ++s) {
      const size_t ls = (size_t)(l * 2 + s);
      gemm(xcur, We + ls * DM * DM, nullptr, nullptr, xe, N, DM, DM);
      gemm(xe, Wq + ls * QKV * DM, nullptr, nullptr, qb, N, DM, QKV);
      gemm(xe, Wk + ls * QKV * DM, nullptr, nullptr, kb, N, DM, QKV);
      gemm(xe, Wv + ls * QKV * DM, nullptr, nullptr, vb, N, DM, QKV);
      int nz = N * QKV;
      zerof<<<(nz + 255) / 256, 256, 0, stream>>>(ab, nz);
      edge_attn<<<(E + 3) / 4, 128, 0, stream>>>(qb, kb, vb, ei, ab, E);
      gemm(ab, Wo + ls * DM * QKV, bo + ls * DM, xe, hb, N, QKV, DM);
      layernorm256<<<N, 256, 0, stream>>>(xcur, hb, ln_g + (l * 3 + s) * DM,
                                          ln_b + (l * 3 + s) * DM, xcur);
    }
    gemm(xcur, mlp_W + (size_t)l * DM * DM, mlp_b + (size_t)l * DM, xcur, hb,
         N, DM, DM);
    float* dest = (l == NLAYERS - 1) ? (float*)d_out : xcur;
    layernorm256<<<N, 256, 0, stream>>>(hb, nullptr, ln_g + (l * 3 + 2) * DM,
                                        ln_b + (l * 3 + 2) * DM, dest);
  }
}